// GPSA_46488726011924
// MI455X (gfx1250) — compile-verified
//
#include <hip/hip_runtime.h>
#include <math.h>

// GPSA forward for MI455X (gfx1250), fp32 WMMA 16x16x4 path.
// Shapes: B=8, N=784 (28x28), DIM=384, H=12, head_dim=32.

#define B_    8
#define N_    784
#define DIM_  384
#define H_    12
#define HD_   32
#define GRID_ 28
#define NT_   49      // N_/16  (16-row tiles of the attention score block)
#define RT64_ 98      // (B_*N_)/64  (64-row GEMM macro-tiles)
#define CT64_ 6       // DIM_/64     (64-col GEMM macro-tiles)
#define SLD_  788     // LDS row stride for S (784 + 4 pad)

typedef float v2f __attribute__((ext_vector_type(2)));
typedef float v8f __attribute__((ext_vector_type(8)));

__device__ __forceinline__ v8f wmma_f32(v2f a, v2f b, v8f c) {
  // D(16x16,f32) = A(16x4,f32) * B(4x16,f32) + C
  return __builtin_amdgcn_wmma_f32_16x16x4_f32(
      /*neg_a=*/false, a, /*neg_b=*/false, b,
      /*c_mod=*/(short)0, c, /*reuse_a=*/false, /*reuse_b=*/false);
}

__device__ __forceinline__ float wave_max(float x) {
#pragma unroll
  for (int off = 16; off > 0; off >>= 1) x = fmaxf(x, __shfl_xor(x, off, 32));
  return x;
}
__device__ __forceinline__ float wave_sum(float x) {
#pragma unroll
  for (int off = 16; off > 0; off >>= 1) x += __shfl_xor(x, off, 32);
  return x;
}

// ---------------------------------------------------------------------------
// Kernel 1: fused Q/K/V projections.  q|k|v = x @ W^T   (6272 x 384 x 384)
// One wave computes a 64x64 output macro-tile as 4x4 WMMA sub-tiles:
// per k-chunk, 4 A + 4 B loads feed 16 WMMAs (0.5 loads/WMMA) with 16
// independent accumulation chains.  Scale H^-0.5 is folded into Q.
// ---------------------------------------------------------------------------
__global__ __launch_bounds__(128) void gpsa_qkv_kernel(
    const float* __restrict__ x,
    const float* __restrict__ Wq, const float* __restrict__ Wk,
    const float* __restrict__ Wv,
    float* __restrict__ qo, float* __restrict__ ko, float* __restrict__ vo) {
  const int w = blockIdx.x * 4 + (threadIdx.x >> 5);
  const int L = threadIdx.x & 31;
  const int ct  = w % CT64_;
  const int rt  = (w / CT64_) % RT64_;
  const int mat = w / (CT64_ * RT64_);   // 0=Q 1=K 2=V

  const float* W = (mat == 0) ? Wq : (mat == 1) ? Wk : Wv;
  float*       O = (mat == 0) ? qo : (mat == 1) ? ko : vo;

  const int g = rt * 64, i0 = ct * 64;
  const int nn = L & 15, koff = (L >> 4) << 1, rowadd = (L >> 4) << 3;

  const float* xa[4];
  const float* wb[4];
#pragma unroll
  for (int i = 0; i < 4; ++i) {
    xa[i] = x + (size_t)(g + 16 * i + nn) * DIM_ + koff;
    wb[i] = W + (size_t)(i0 + 16 * i + nn) * DIM_ + koff;
  }

  v8f acc[4][4];
#pragma unroll
  for (int i = 0; i < 4; ++i)
#pragma unroll
    for (int j = 0; j < 4; ++j)
      acc[i][j] = (v8f){0.f, 0.f, 0.f, 0.f, 0.f, 0.f, 0.f, 0.f};

#pragma unroll 2
  for (int k4 = 0; k4 < DIM_; k4 += 4) {
    v2f a[4], b[4];
#pragma unroll
    for (int i = 0; i < 4; ++i) a[i] = *(const v2f*)(xa[i] + k4);
#pragma unroll
    for (int j = 0; j < 4; ++j) b[j] = *(const v2f*)(wb[j] + k4);
#pragma unroll
    for (int i = 0; i < 4; ++i)
#pragma unroll
      for (int j = 0; j < 4; ++j) acc[i][j] = wmma_f32(a[i], b[j], acc[i][j]);
  }

  const float sc = (mat == 0) ? 0.28867513459481287f : 1.0f;  // 12^-0.5
#pragma unroll
  for (int i = 0; i < 4; ++i)
#pragma unroll
    for (int j = 0; j < 4; ++j)
#pragma unroll
      for (int r = 0; r < 8; ++r)
        O[(size_t)(g + 16 * i + r + rowadd) * DIM_ + i0 + 16 * j + nn] =
            acc[i][j][r] * sc;
}

// ---------------------------------------------------------------------------
// Kernel 2: positional-softmax row statistics.
// pos_logit(h,n,m) = Wp0*dx + Wp1*dy + Wp2*(dx^2+dy^2) + b_pos[h],
// dx = m%28 - n%28, dy = m/28 - n/28.  Store rowmax and 1/rowsum(exp).
// One wave per (h,n) row.
// ---------------------------------------------------------------------------
__global__ __launch_bounds__(128) void gpsa_pos_stats_kernel(
    const float* __restrict__ W_pos, const float* __restrict__ b_pos,
    float* __restrict__ pmx, float* __restrict__ pinv) {
  const int w = blockIdx.x * 4 + (threadIdx.x >> 5);
  const int L = threadIdx.x & 31;
  const int h = w / N_, n = w % N_;
  const float wp0 = W_pos[h * 3 + 0], wp1 = W_pos[h * 3 + 1],
              wp2 = W_pos[h * 3 + 2], bp = b_pos[h];
  const int nx = n % GRID_, ny = n / GRID_;

  float mx = -3.4e38f;
  for (int m = L; m < N_; m += 32) {
    float dx = (float)(m % GRID_ - nx), dy = (float)(m / GRID_ - ny);
    float lg = wp0 * dx + wp1 * dy + wp2 * (dx * dx + dy * dy) + bp;
    mx = fmaxf(mx, lg);
  }
  mx = wave_max(mx);
  float s = 0.f;
  for (int m = L; m < N_; m += 32) {
    float dx = (float)(m % GRID_ - nx), dy = (float)(m / GRID_ - ny);
    float lg = wp0 * dx + wp1 * dy + wp2 * (dx * dx + dy * dy) + bp;
    s += __expf(lg - mx);
  }
  s = wave_sum(s);
  if (L == 0) { pmx[w] = mx; pinv[w] = 1.f / s; }
}

// ---------------------------------------------------------------------------
// Kernel 3: attention core.  One wave per (b, h, 16-row n-tile).
//   Pass A: S[16,784] = Q * K^T into LDS; Q operands hoisted to registers
//   Softmax rows; gated mix with analytic pos softmax; exact renorm
//   Pass B: O[16,32]  = S * V; LDS A-operand shared by both d-tiles,
//           two independent accumulation chains.
// ---------------------------------------------------------------------------
__global__ __launch_bounds__(32) void gpsa_attn_kernel(
    const float* __restrict__ q, const float* __restrict__ k,
    const float* __restrict__ v,
    const float* __restrict__ W_pos, const float* __restrict__ b_pos,
    const float* __restrict__ gating,
    const float* __restrict__ pmx_, const float* __restrict__ pinv_,
    float* __restrict__ o) {
  __shared__ float S[16 * SLD_];   // 50432 B

  const int L = threadIdx.x;
  const int task = blockIdx.x;           // b*H*NT + h*NT + nt
  const int nt = task % NT_;
  const int h  = (task / NT_) % H_;
  const int b  = task / (NT_ * H_);
  const int n0 = nt * 16;
  const int nn = L & 15, koff = (L >> 4) << 1, rowadd = (L >> 4) << 3;

  const float* Qb = q + (size_t)b * N_ * DIM_ + h * HD_;
  const float* Kb = k + (size_t)b * N_ * DIM_ + h * HD_;
  const float* Vb = v + (size_t)b * N_ * DIM_ + h * HD_;

  // ---- Pass A: S = Q K^T  (scale already folded into Q) ----
  // Q A-operands are invariant across m-tiles: keep all 8 in registers.
  const float* qa = Qb + (size_t)(n0 + nn) * DIM_ + koff;
  v2f aop[8];
#pragma unroll
  for (int i = 0; i < 8; ++i) aop[i] = *(const v2f*)(qa + 4 * i);

  for (int mt = 0; mt < NT_; ++mt) {
    const int m0 = mt * 16;
    const float* kb = Kb + (size_t)(m0 + nn) * DIM_ + koff;
    v8f acc = {0.f, 0.f, 0.f, 0.f, 0.f, 0.f, 0.f, 0.f};
#pragma unroll
    for (int i = 0; i < 8; ++i) {
      v2f bb = *(const v2f*)(kb + 4 * i);
      acc = wmma_f32(aop[i], bb, acc);
    }
#pragma unroll
    for (int r = 0; r < 8; ++r) S[(r + rowadd) * SLD_ + m0 + nn] = acc[r];
  }
  __syncthreads();

  // ---- Softmax + gated positional mix + renorm ----
  const float gh = gating[h];
  const float gs = 1.f / (1.f + __expf(-gh));
  const float c1 = 1.f - gs;
  const float wp0 = W_pos[h * 3 + 0], wp1 = W_pos[h * 3 + 1],
              wp2 = W_pos[h * 3 + 2], bp = b_pos[h];

  for (int r = 0; r < 16; ++r) {
    const int n = n0 + r;
    float* Sr = &S[r * SLD_];
    float mx = -3.4e38f;
    for (int m = L; m < N_; m += 32) mx = fmaxf(mx, Sr[m]);
    mx = wave_max(mx);
    float s = 0.f;
    for (int m = L; m < N_; m += 32) {
      float p = __expf(Sr[m] - mx);
      Sr[m] = p;
      s += p;
    }
    s = wave_sum(s);
    const float rinv = 1.f / s;
    const float pm = pmx_[h * N_ + n], pv = pinv_[h * N_ + n];
    const int nx = n % GRID_, ny = n / GRID_;
    float rs = 0.f;
    for (int m = L; m < N_; m += 32) {
      float dx = (float)(m % GRID_ - nx), dy = (float)(m / GRID_ - ny);
      float lg = wp0 * dx + wp1 * dy + wp2 * (dx * dx + dy * dy) + bp;
      float pp = __expf(lg - pm) * pv;
      float t = c1 * Sr[m] * rinv + gs * pp;
      Sr[m] = t;
      rs += t;
    }
    rs = wave_sum(rs);
    const float rn = 1.f / rs;        // ~1.0; kept for exact fidelity
    for (int m = L; m < N_; m += 32) Sr[m] *= rn;
  }
  __syncthreads();

  // ---- Pass B: O = S * V  (both 16-col d-tiles per m-chunk) ----
  v8f acc0 = {0.f, 0.f, 0.f, 0.f, 0.f, 0.f, 0.f, 0.f};
  v8f acc1 = {0.f, 0.f, 0.f, 0.f, 0.f, 0.f, 0.f, 0.f};
#pragma unroll 4
  for (int mc = 0; mc < N_ / 4; ++mc) {
    const int m0 = mc * 4;
    v2f a = *(const v2f*)&S[nn * SLD_ + m0 + koff];
    const float* vb = Vb + (size_t)(m0 + koff) * DIM_ + nn;
    v2f b0 = {vb[0], vb[DIM_]};
    v2f b1 = {vb[16], vb[DIM_ + 16]};
    acc0 = wmma_f32(a, b0, acc0);
    acc1 = wmma_f32(a, b1, acc1);
  }
  const int col = h * HD_ + nn;
#pragma unroll
  for (int r = 0; r < 8; ++r) {
    const size_t row = (size_t)b * N_ + n0 + r + rowadd;
    o[row * DIM_ + col]      = acc0[r];
    o[row * DIM_ + col + 16] = acc1[r];
  }
}

// ---------------------------------------------------------------------------
// Kernel 4: output projection.  out = o_tmp @ W_out^T + b_out
// Same 64x64 register-blocked structure as the QKV GEMM.
// ---------------------------------------------------------------------------
__global__ __launch_bounds__(128) void gpsa_proj_kernel(
    const float* __restrict__ xin, const float* __restrict__ Wo,
    const float* __restrict__ bo, float* __restrict__ out) {
  const int w = blockIdx.x * 4 + (threadIdx.x >> 5);
  const int L = threadIdx.x & 31;
  const int ct = w % CT64_;
  const int rt = w / CT64_;
  const int g = rt * 64, i0 = ct * 64;
  const int nn = L & 15, koff = (L >> 4) << 1, rowadd = (L >> 4) << 3;

  const float* xa[4];
  const float* wb[4];
#pragma unroll
  for (int i = 0; i < 4; ++i) {
    xa[i] = xin + (size_t)(g + 16 * i + nn) * DIM_ + koff;
    wb[i] = Wo + (size_t)(i0 + 16 * i + nn) * DIM_ + koff;
  }

  v8f acc[4][4];
#pragma unroll
  for (int i = 0; i < 4; ++i)
#pragma unroll
    for (int j = 0; j < 4; ++j)
      acc[i][j] = (v8f){0.f, 0.f, 0.f, 0.f, 0.f, 0.f, 0.f, 0.f};

#pragma unroll 2
  for (int k4 = 0; k4 < DIM_; k4 += 4) {
    v2f a[4], b[4];
#pragma unroll
    for (int i = 0; i < 4; ++i) a[i] = *(const v2f*)(xa[i] + k4);
#pragma unroll
    for (int j = 0; j < 4; ++j) b[j] = *(const v2f*)(wb[j] + k4);
#pragma unroll
    for (int i = 0; i < 4; ++i)
#pragma unroll
      for (int j = 0; j < 4; ++j) acc[i][j] = wmma_f32(a[i], b[j], acc[i][j]);
  }

#pragma unroll
  for (int j = 0; j < 4; ++j) {
    const float bias = bo[i0 + 16 * j + nn];
#pragma unroll
    for (int i = 0; i < 4; ++i)
#pragma unroll
      for (int r = 0; r < 8; ++r)
        out[(size_t)(g + 16 * i + r + rowadd) * DIM_ + i0 + 16 * j + nn] =
            acc[i][j][r] + bias;
  }
}

// ---------------------------------------------------------------------------
extern "C" void kernel_launch(void* const* d_in, const int* in_sizes, int n_in,
                              void* d_out, int out_size, void* d_ws,
                              size_t ws_size, hipStream_t stream) {
  (void)in_sizes; (void)n_in; (void)out_size; (void)ws_size;
  const float* x      = (const float*)d_in[0];
  const float* Wq     = (const float*)d_in[1];
  const float* Wk     = (const float*)d_in[2];
  const float* Wv     = (const float*)d_in[3];
  const float* W_pos  = (const float*)d_in[4];
  const float* b_pos  = (const float*)d_in[5];
  const float* W_out  = (const float*)d_in[6];
  const float* b_out  = (const float*)d_in[7];
  const float* gating = (const float*)d_in[8];
  float* out = (float*)d_out;

  // Workspace layout (floats): q | k | v | o_tmp | pos_max | pos_invsum
  const size_t MAT = (size_t)B_ * N_ * DIM_;   // 2,408,448
  float* ws   = (float*)d_ws;
  float* qws  = ws;
  float* kws  = ws + MAT;
  float* vws  = ws + 2 * MAT;
  float* ows  = ws + 3 * MAT;
  float* pmx  = ws + 4 * MAT;                  // H_*N_ = 9408
  float* pinv = pmx + (size_t)H_ * N_;

  // 1) QKV projections: 3 * 98 * 6 = 1764 wave-tiles (64x64), 4 waves/block
  gpsa_qkv_kernel<<<(3 * RT64_ * CT64_) / 4, 128, 0, stream>>>(
      x, Wq, Wk, Wv, qws, kws, vws);

  // 2) Positional softmax stats: H*N waves
  gpsa_pos_stats_kernel<<<(H_ * N_) / 4, 128, 0, stream>>>(
      W_pos, b_pos, pmx, pinv);

  // 3) Attention core: B*H*49 single-wave blocks
  gpsa_attn_kernel<<<B_ * H_ * NT_, 32, 0, stream>>>(
      qws, kws, vws, W_pos, b_pos, gating, pmx, pinv, ows);

  // 4) Output projection + bias: 98 * 6 = 588 wave-tiles (64x64)
  gpsa_proj_kernel<<<(RT64_ * CT64_) / 4, 128, 0, stream>>>(
      ows, W_out, b_out, out);
}